// CNNFocusing_1614907703356
// MI455X (gfx1250) — compile-verified
//
#include <hip/hip_runtime.h>
#include <hip/hip_bf16.h>
#include <cstdint>

// ---- problem constants (from reference) ----
#define BATCH   128
#define GRP     72
#define S1      32          // conv1 spatial
#define S2      16          // conv2 spatial
#define HD      1536        // hidden per landmark = 6*16*16
#define HPL     10          // hidden_per_lmark
#define W1_STRIDE (GRP * HD)    // 110592 (dense W1 row stride)
#define W2_STRIDE (GRP * HPL)   // 720    (dense W2 row stride)

typedef float v2f __attribute__((ext_vector_type(2)));
typedef float v8f __attribute__((ext_vector_type(8)));
typedef unsigned int v4u __attribute__((ext_vector_type(4)));
typedef int v4i __attribute__((ext_vector_type(4)));
typedef int v8i __attribute__((ext_vector_type(8)));

// =====================================================================
// Kernel 1: fused conv1 + ReLU + maxpool2 + conv2 + ReLU per (group,batch)
// Writes h2 laid out [B, G, 1536] (matches NCHW flatten -> (B, G, 1536)).
// =====================================================================
__global__ __launch_bounds__(256) void conv_fused(
    const float* __restrict__ x,      // [B,216,32,32]
    const float* __restrict__ cw1,    // [216,3,3,3]
    const float* __restrict__ cb1,    // [216]
    const float* __restrict__ cw2,    // [432,3,3,3]
    const float* __restrict__ cb2,    // [432]
    float* __restrict__ h2)           // [B,G,1536]
{
    const int g   = blockIdx.x;
    const int b   = blockIdx.y;
    const int tid = threadIdx.x;

    __shared__ float xs[3][34][34];   // edge-padded input tile
    __shared__ float p1[3][18][18];   // edge-padded pooled conv1 output
    __shared__ float w1s[3][27];      // conv1 weights for this group
    __shared__ float w2s[6][27];      // conv2 weights for this group
    __shared__ float b1s[3];
    __shared__ float b2s[6];

    // stage weights / biases
    if (tid < 81) {
        int oc = tid / 27, r = tid % 27;
        w1s[oc][r] = cw1[(g * 3 + oc) * 27 + r];
    } else if (tid < 243) {
        int t = tid - 81;
        int oc = t / 27, r = t % 27;
        w2s[oc][r] = cw2[(g * 6 + oc) * 27 + r];
    } else if (tid < 246) {
        b1s[tid - 243] = cb1[g * 3 + (tid - 243)];
    } else if (tid < 252) {
        b2s[tid - 246] = cb2[g * 6 + (tid - 246)];
    }

    // stage input with edge ('same'/replicate) padding
    const float* xg = x + ((size_t)b * 216 + g * 3) * (S1 * S1);
    for (int i = tid; i < 3 * 34 * 34; i += 256) {
        int c  = i / 1156;
        int r  = i % 1156;
        int py = r / 34, px = r % 34;
        int gy = min(max(py - 1, 0), S1 - 1);
        int gx = min(max(px - 1, 0), S1 - 1);
        xs[c][py][px] = xg[(c * S1 + gy) * S1 + gx];
    }
    __syncthreads();

    // conv1 + bias + ReLU + 2x2 maxpool -> p1 interior ([1..16])
    for (int i = tid; i < 3 * 256; i += 256) {
        int c  = i >> 8;
        int yy = (i >> 4) & 15;
        int xx = i & 15;
        float m = 0.f;                         // relu(max(..)) == max(0, max(..))
        #pragma unroll
        for (int dy = 0; dy < 2; ++dy)
        #pragma unroll
        for (int dx = 0; dx < 2; ++dx) {
            int Y = 2 * yy + dy, X = 2 * xx + dx;
            float acc = b1s[c];
            #pragma unroll
            for (int ic = 0; ic < 3; ++ic)
            #pragma unroll
            for (int ky = 0; ky < 3; ++ky)
            #pragma unroll
            for (int kx = 0; kx < 3; ++kx)
                acc += xs[ic][Y + ky][X + kx] * w1s[c][ic * 9 + ky * 3 + kx];
            m = fmaxf(m, acc);
        }
        p1[c][yy + 1][xx + 1] = m;
    }
    __syncthreads();

    // replicate edges of the pooled map (reads interior only, writes ring only)
    for (int i = tid; i < 3 * 18 * 18; i += 256) {
        int c = i / 324, r = i % 324;
        int py = r / 18, px = r % 18;
        if (py == 0 || py == 17 || px == 0 || px == 17)
            p1[c][py][px] = p1[c][min(max(py, 1), 16)][min(max(px, 1), 16)];
    }
    __syncthreads();

    // conv2 + bias + ReLU -> h2 (coalesced: consecutive tid -> consecutive k)
    float* out = h2 + ((size_t)b * GRP + g) * HD;
    for (int i = tid; i < HD; i += 256) {
        int c  = i >> 8;            // 0..5
        int yy = (i >> 4) & 15;
        int xx = i & 15;
        float acc = b2s[c];
        #pragma unroll
        for (int ic = 0; ic < 3; ++ic)
        #pragma unroll
        for (int ky = 0; ky < 3; ++ky)
        #pragma unroll
        for (int kx = 0; kx < 3; ++kx)
            acc += p1[ic][yy + ky][xx + kx] * w2s[c][ic * 9 + ky * 3 + kx];
        out[i] = fmaxf(acc, 0.f);
    }
}

// =====================================================================
// Kernel 2: extract the 72 diagonal blocks of W1 into a packed buffer
// w1x[g][k][o16] with o zero-padded 10->16 (ready-made WMMA B panels).
// =====================================================================
__global__ __launch_bounds__(256) void extract_w1(
    const float* __restrict__ W1,   // [720, 110592]
    float* __restrict__ w1x)        // [G, 1536, 16]
{
    int i = blockIdx.x * 256 + threadIdx.x;
    if (i >= GRP * HD * 16) return;
    int g = i / (HD * 16);
    int r = i % (HD * 16);
    int k = r >> 4;
    int o = r & 15;
    w1x[i] = (o < HPL)
           ? W1[(size_t)(g * HPL + o) * W1_STRIDE + (size_t)g * HD + k]
           : 0.f;
}

// =====================================================================
// TDM helper: issue a 2D tile load (global -> LDS) via the Tensor Data
// Mover.  D# layout per CDNA5 ISA ch.8: group0 = {flags, lds_addr,
// global_addr, type=2}; group1 = {dma cfg, tensor dims, tile dims,
// dim0 stride}.  Groups 2/3 (and the trailing extension group) are zero
// for a <=2D tensor.  All inputs uniform -> descriptors live in SGPRs.
// This toolchain's builtin is the 6-arg clang-23 form:
//   (uint32x4 g0, int32x8 g1, int32x4 g2, int32x4 g3, int32x8 gx, i32 cpol)
// =====================================================================
__device__ __forceinline__ void tdm_load_2d(
    uint32_t lds_off,            // LDS byte offset of tile destination
    const float* gaddr,          // global tile start
    uint32_t tile_d0,            // tile width  (elements)
    uint32_t tile_d1,            // tile height (rows)
    uint32_t stride0,            // row stride  (elements)
    uint32_t g1w0)               // group1 dword0: data_size / pad config
{
    uint64_t ga = (uint64_t)(uintptr_t)gaddr;
    v4u g0;
    g0.x = 1u;                                   // count=1, user descriptor
    g0.y = lds_off;                              // lds_addr
    g0.z = (uint32_t)ga;                         // global_addr[31:0]
    g0.w = (uint32_t)(ga >> 32) | (2u << 30);    // global_addr[56:32] | type=2
    v8i g1;
    g1[0] = (int)g1w0;                                            // mask=0|data_size|pad
    g1[1] = (int)((tile_d0 & 0xffffu) << 16);                     // tensor_dim0 = tile_d0
    g1[2] = (int)((tile_d0 >> 16) | ((tile_d1 & 0xffffu) << 16)); // tensor_dim1 = tile_d1
    g1[3] = (int)((tile_d1 >> 16) | (tile_d0 << 16));             // tile_dim0
    g1[4] = (int)(tile_d1 & 0xffffu);                             // tile_dim1 (tile_dim2=0)
    g1[5] = (int)stride0;                                         // tensor_dim0_stride lo32
    g1[6] = 0;                                                    // stride0 hi, stride1 lo
    g1[7] = 0;
    v4i z4 = {0, 0, 0, 0};
    v8i z8 = {0, 0, 0, 0, 0, 0, 0, 0};
    __builtin_amdgcn_tensor_load_to_lds(g0, g1, z4, z4, z8, 0);
}

// group1 dword0 encodings (data_size=2 -> 4-byte elements)
//  A panel: pad_enable, pad_interval=4 (32 DW), pad_amount=1 (2 DW)
//           -> LDS row stride 34 floats, matching As[16][34]
#define G1W0_A 0x03120000u
//  B panel: plain contiguous copy
#define G1W0_B 0x00020000u

// =====================================================================
// Kernel 3: block-diagonal linear1 as WMMA f32 GEMM (M=16,N=16,K=1536)
// per (group, batch-tile).  TDM double-buffered LDS pipeline; bias in
// the accumulator; linear2 (10 -> 2) fused in the epilogue.
// One wave per block (TDM + WMMA are wave-level ops).
// =====================================================================
__global__ __launch_bounds__(32) void gemm_head(
    const float* __restrict__ h2,    // [B, G, 1536]
    const float* __restrict__ w1x,   // [G, 1536, 16]
    const float* __restrict__ b1,    // [720]
    const float* __restrict__ W2,    // [144, 720]
    const float* __restrict__ b2,    // [144]
    float* __restrict__ out)         // [B, 144]
{
    const int bt   = blockIdx.x;       // batch tile (16 rows)
    const int g    = blockIdx.y;       // landmark group
    const int lane = threadIdx.x;      // wave32
    const int l16  = lane & 15;        // M index (A) / N index (B,C,D)
    const int kh   = (lane >> 4) << 1; // K-pair selector: 0 or 2

    __shared__ float As[2][16][34];    // TDM pads rows 32 -> 34 floats
    __shared__ float Bs[2][32][16];
    __shared__ float Ys[16][12];

    // C init = per-column bias (broadcast down all 16 rows of the tile)
    v8f acc;
    {
        float bias = (l16 < HPL) ? b1[g * HPL + l16] : 0.f;
        #pragma unroll
        for (int r = 0; r < 8; ++r) acc[r] = bias;
    }

    const float* a_base = h2 + ((size_t)(bt * 16) * GRP + g) * HD; // row stride G*HD
    const float* b_base = w1x + (size_t)g * HD * 16;

    const uint32_t asOff    = (uint32_t)(uintptr_t)&As[0][0][0];
    const uint32_t bsOff    = (uint32_t)(uintptr_t)&Bs[0][0][0];
    const uint32_t asStride = sizeof(float) * 16 * 34;
    const uint32_t bsStride = sizeof(float) * 32 * 16;

    // prologue: chunk 0 -> buffer 0 (A: 16x32 strided, B: 512x1 contiguous)
    tdm_load_2d(asOff, a_base, 32, 16, GRP * HD, G1W0_A);
    tdm_load_2d(bsOff, b_base, 512, 1, 512, G1W0_B);

    for (int k0 = 0; k0 < HD; k0 += 32) {
        const int cur = (k0 >> 5) & 1;
        if (k0 + 32 < HD) {
            const int nxt = cur ^ 1;
            // make sure no ds reads of the nxt buffer are still in flight
            asm volatile("s_wait_dscnt 0x0" ::: "memory");
            tdm_load_2d(asOff + nxt * asStride, a_base + (k0 + 32), 32, 16,
                        GRP * HD, G1W0_A);
            tdm_load_2d(bsOff + nxt * bsStride, b_base + ((k0 + 32) << 4), 512, 1,
                        512, G1W0_B);
            // TDM completes in-order: <=2 outstanding => current pair landed
            __builtin_amdgcn_s_wait_tensorcnt(2);
        } else {
            __builtin_amdgcn_s_wait_tensorcnt(0);
        }

        #pragma unroll
        for (int kk = 0; kk < 32; kk += 4) {
            // A 16x4 f32 frag: lane holds A[l16][kk+kh .. kk+kh+1]
            v2f a = *(const v2f*)&As[cur][l16][kk + kh];
            // B 4x16 f32 frag: lane holds B[kk+kh .. kk+kh+1][l16]
            v2f bf;
            bf.x = Bs[cur][kk + kh + 0][l16];
            bf.y = Bs[cur][kk + kh + 1][l16];
            acc = __builtin_amdgcn_wmma_f32_16x16x4_f32(
                /*neg_a=*/false, a, /*neg_b=*/false, bf,
                /*c_mod=*/(short)0, acc, /*reuse_a=*/false, /*reuse_b=*/false);
        }
    }

    // spill y1 tile (16 batches x 10 outputs) to LDS
    if (l16 < HPL) {
        const int mo = (lane >> 4) << 3;   // rows 0..7 or 8..15
        #pragma unroll
        for (int r = 0; r < 8; ++r) Ys[r + mo][l16] = acc[r];
    }
    __syncthreads();

    // fused linear2: 32 lanes = 16 batches x 2 outputs
    {
        const int m = lane >> 1;
        const int c = lane & 1;
        const float* w2r = W2 + (size_t)(g * 2 + c) * W2_STRIDE + g * HPL;
        float v = b2[g * 2 + c];
        #pragma unroll
        for (int o = 0; o < HPL; ++o) v += Ys[m][o] * w2r[o];
        out[(size_t)(bt * 16 + m) * (GRP * 2) + g * 2 + c] = v;
    }
}

// =====================================================================
extern "C" void kernel_launch(void* const* d_in, const int* in_sizes, int n_in,
                              void* d_out, int out_size, void* d_ws, size_t ws_size,
                              hipStream_t stream) {
    (void)in_sizes; (void)n_in; (void)out_size; (void)ws_size;

    const float* x   = (const float*)d_in[0];
    const float* cw1 = (const float*)d_in[1];
    const float* cb1 = (const float*)d_in[2];
    const float* cw2 = (const float*)d_in[3];
    const float* cb2 = (const float*)d_in[4];
    const float* W1  = (const float*)d_in[5];
    const float* b1  = (const float*)d_in[6];
    const float* W2  = (const float*)d_in[7];
    const float* b2  = (const float*)d_in[8];
    float* out = (float*)d_out;

    // workspace carve-up (floats)
    float* h2  = (float*)d_ws;                                  // B*G*HD  = 14,155,776
    float* w1x = h2 + (size_t)BATCH * GRP * HD;                 // G*HD*16 =  1,769,472

    // 1) fused convs: one block per (group, batch)
    conv_fused<<<dim3(GRP, BATCH), 256, 0, stream>>>(x, cw1, cb1, cw2, cb2, h2);

    // 2) pack W1 diagonal blocks
    {
        int total = GRP * HD * 16;
        extract_w1<<<(total + 255) / 256, 256, 0, stream>>>(W1, w1x);
    }

    // 3) TDM + WMMA GEMM + fused linear2: one wave per (batch-tile, group)
    gemm_head<<<dim3(BATCH / 16, GRP), 32, 0, stream>>>(h2, w1x, b1, W2, b2, out);
}